// VDM_data_84748294684882
// MI455X (gfx1250) — compile-verified
//
#include <hip/hip_runtime.h>
#include <math.h>

typedef float v2f __attribute__((ext_vector_type(2)));
typedef float v4f __attribute__((ext_vector_type(4)));
typedef float v8f __attribute__((ext_vector_type(8)));

#define IMG_DIM 3072
#define NDB     10000
#define NB      16

__device__ __forceinline__ v2f lo2(v4f v) { return __builtin_shufflevector(v, v, 0, 1); }
__device__ __forceinline__ v2f hi2(v4f v) { return __builtin_shufflevector(v, v, 2, 3); }

// ---------------------------------------------------------------------------
// Kernel A: fimg[b,k] = images[b,k]*sigma[b] + db[0,k]*alpha[b]
//           xnorm[b]  = ||fimg_b||^2 ; den[b] = 0
// ---------------------------------------------------------------------------
__global__ __launch_bounds__(256) void prep_kernel(
    const float* __restrict__ db, const float* __restrict__ sigma,
    const float* __restrict__ alpha, const float* __restrict__ images,
    float* __restrict__ fimg, float* __restrict__ xnorm, float* __restrict__ den)
{
  const int b = blockIdx.x;
  const int t = threadIdx.x;
  const float s = sigma[b], a = alpha[b];
  float acc = 0.f;
  for (int k = t; k < IMG_DIM; k += 256) {
    float x = fmaf(images[b * IMG_DIM + k], s, db[k] * a);
    fimg[b * IMG_DIM + k] = x;
    acc = fmaf(x, x, acc);
  }
  __shared__ float red[256];
  red[t] = acc;
  __syncthreads();
  for (int off = 128; off > 0; off >>= 1) {
    if (t < off) red[t] += red[t + off];
    __syncthreads();
  }
  if (t == 0) { xnorm[b] = red[0]; den[b] = 0.f; }
}

// ---------------------------------------------------------------------------
// Kernel B: S[b,n] = fimg_b . db_n  via V_WMMA_F32_16X16X4_F32  (K = 3072)
//   8 waves/block, one 16-row db tile per block, waves split K (384 each).
//   Per iteration: 2 x b128 loads/lane -> 2 WMMAs (K-permutation trick).
//   2-deep software prefetch pipeline.
// ---------------------------------------------------------------------------
__global__ __launch_bounds__(256) void dist_pdf_kernel(
    const float* __restrict__ db, const float* __restrict__ sigma,
    const float* __restrict__ alpha, const float* __restrict__ fimg,
    const float* __restrict__ xnorm, float* __restrict__ pdfbuf,
    float* __restrict__ den)
{
  const int tid  = threadIdx.x;
  const int wave = tid >> 5;
  const int lane = tid & 31;
  const int half = lane >> 4;
  const int l15  = lane & 15;
  const int n0   = blockIdx.x * 16;

  __shared__ float sS[8][16][16];
  __shared__ float sdn[16];
  __shared__ float sden[16];
  if (tid < 16) { sdn[tid] = 0.f; sden[tid] = 0.f; }
  __syncthreads();

  // lane j<16 owns k..k+3, lane j+16 owns k+4..k+7 of row j (same map for A & B)
  const float* arow = fimg + (size_t)l15 * IMG_DIM + half * 4;
  const float* brow = db + (size_t)(n0 + l15) * IMG_DIM + half * 4;

  v8f acc = {};
  float dn = 0.f;
  const int kbeg = wave * (IMG_DIM / 8);       // 384 K-values per wave
  const int kend = kbeg + (IMG_DIM / 8);

  v4f a0 = *(const v4f*)(arow + kbeg);
  v4f b0 = *(const v4f*)(brow + kbeg);
  v4f a1 = *(const v4f*)(arow + kbeg + 8);
  v4f b1 = *(const v4f*)(brow + kbeg + 8);

  for (int k = kbeg; k < kend - 16; k += 8) {
    v4f a2 = *(const v4f*)(arow + k + 16);
    v4f b2 = *(const v4f*)(brow + k + 16);
    dn = fmaf(b0.x, b0.x, dn); dn = fmaf(b0.y, b0.y, dn);
    dn = fmaf(b0.z, b0.z, dn); dn = fmaf(b0.w, b0.w, dn);
    acc = __builtin_amdgcn_wmma_f32_16x16x4_f32(false, lo2(a0), false, lo2(b0),
                                                (short)0, acc, false, false);
    acc = __builtin_amdgcn_wmma_f32_16x16x4_f32(false, hi2(a0), false, hi2(b0),
                                                (short)0, acc, false, false);
    a0 = a1; b0 = b1; a1 = a2; b1 = b2;
  }
  // drain the 2-stage pipeline
  for (int e = 0; e < 2; ++e) {
    dn = fmaf(b0.x, b0.x, dn); dn = fmaf(b0.y, b0.y, dn);
    dn = fmaf(b0.z, b0.z, dn); dn = fmaf(b0.w, b0.w, dn);
    acc = __builtin_amdgcn_wmma_f32_16x16x4_f32(false, lo2(a0), false, lo2(b0),
                                                (short)0, acc, false, false);
    acc = __builtin_amdgcn_wmma_f32_16x16x4_f32(false, hi2(a0), false, hi2(b0),
                                                (short)0, acc, false, false);
    a0 = a1; b0 = b1;
  }

  atomicAdd(&sdn[l15], dn);
#pragma unroll
  for (int r = 0; r < 8; ++r) sS[wave][r + half * 8][l15] = acc[r];
  __syncthreads();

  if (tid < 256) {
    const int bb = tid >> 4, nn = tid & 15;
    float s = 0.f;
#pragma unroll
    for (int w = 0; w < 8; ++w) s += sS[w][bb][nn];
    float al = alpha[bb], sg = sigma[bb];
    float upper = (xnorm[bb] - 2.f * al * s + al * al * sdn[nn]) /
                  (2.f * sg * sg * (float)IMG_DIM);
    float p = __expf(-upper);
    pdfbuf[(size_t)bb * NDB + n0 + nn] = p;
    atomicAdd(&sden[bb], p);
  }
  __syncthreads();
  if (tid < 16) atomicAdd(&den[tid], sden[tid]);
}

// ---------------------------------------------------------------------------
// Kernel C: out[b,j] = (sum_n pdf[b,n]*db[n,j]) / den[b]
//   GEMM M=16 (b), N=3072 (16-wide j tiles), K=10000 (n), fp32 WMMA.
//   8 waves/block split K: 8*1248 + 16-element tail (4 step-4 WMMAs, waves 0-3).
// ---------------------------------------------------------------------------
__global__ __launch_bounds__(256) void wsum_kernel(
    const float* __restrict__ db, const float* __restrict__ pdfbuf,
    const float* __restrict__ den, float* __restrict__ out)
{
  const int tid  = threadIdx.x;
  const int wave = tid >> 5;
  const int lane = tid & 31;
  const int half = lane >> 4;
  const int l15  = lane & 15;
  const int j0   = blockIdx.x * 16;

  __shared__ float sS[8][16][16];

  const float* arow = pdfbuf + (size_t)l15 * NDB + half * 4;
  const float* bcol = db + j0 + l15;

  v8f acc = {};
  const int kbeg = wave * 1248;
  const int kend = kbeg + 1248;

  v4f a0 = *(const v4f*)(arow + kbeg);
  v4f b0, b1;
  {
    const size_t r = (size_t)(kbeg + half * 4) * IMG_DIM;
    b0.x = bcol[r]; b0.y = bcol[r + IMG_DIM];
    b0.z = bcol[r + 2 * (size_t)IMG_DIM]; b0.w = bcol[r + 3 * (size_t)IMG_DIM];
  }
  v4f a1 = *(const v4f*)(arow + kbeg + 8);
  {
    const size_t r = (size_t)(kbeg + 8 + half * 4) * IMG_DIM;
    b1.x = bcol[r]; b1.y = bcol[r + IMG_DIM];
    b1.z = bcol[r + 2 * (size_t)IMG_DIM]; b1.w = bcol[r + 3 * (size_t)IMG_DIM];
  }

  for (int k = kbeg; k < kend - 16; k += 8) {
    v4f a2 = *(const v4f*)(arow + k + 16);
    v4f b2;
    const size_t r = (size_t)(k + 16 + half * 4) * IMG_DIM;
    b2.x = bcol[r]; b2.y = bcol[r + IMG_DIM];
    b2.z = bcol[r + 2 * (size_t)IMG_DIM]; b2.w = bcol[r + 3 * (size_t)IMG_DIM];
    acc = __builtin_amdgcn_wmma_f32_16x16x4_f32(false, lo2(a0), false, lo2(b0),
                                                (short)0, acc, false, false);
    acc = __builtin_amdgcn_wmma_f32_16x16x4_f32(false, hi2(a0), false, hi2(b0),
                                                (short)0, acc, false, false);
    a0 = a1; b0 = b1; a1 = a2; b1 = b2;
  }
  for (int e = 0; e < 2; ++e) {
    acc = __builtin_amdgcn_wmma_f32_16x16x4_f32(false, lo2(a0), false, lo2(b0),
                                                (short)0, acc, false, false);
    acc = __builtin_amdgcn_wmma_f32_16x16x4_f32(false, hi2(a0), false, hi2(b0),
                                                (short)0, acc, false, false);
    a0 = a1; b0 = b1;
  }

  // K tail: 16 values at 9984..9999, four step-4 WMMAs on waves 0..3
  if (wave < 4) {
    const int kt = 8 * 1248 + wave * 4;
    v2f at = *(const v2f*)(pdfbuf + (size_t)l15 * NDB + kt + half * 2);
    v2f bt;
    const size_t r = (size_t)(kt + half * 2) * IMG_DIM;
    bt.x = bcol[r]; bt.y = bcol[r + IMG_DIM];
    acc = __builtin_amdgcn_wmma_f32_16x16x4_f32(false, at, false, bt,
                                                (short)0, acc, false, false);
  }

#pragma unroll
  for (int r = 0; r < 8; ++r) sS[wave][r + half * 8][l15] = acc[r];
  __syncthreads();

  {
    const int bb = tid >> 4, nn = tid & 15;
    float s = 0.f;
#pragma unroll
    for (int w = 0; w < 8; ++w) s += sS[w][bb][nn];
    out[(size_t)bb * IMG_DIM + j0 + nn] = s / den[bb];
  }
}

// ---------------------------------------------------------------------------
extern "C" void kernel_launch(void* const* d_in, const int* in_sizes, int n_in,
                              void* d_out, int out_size, void* d_ws, size_t ws_size,
                              hipStream_t stream)
{
  (void)in_sizes; (void)n_in; (void)out_size; (void)ws_size;
  const float* db     = (const float*)d_in[0];  // (1,10000,3072)
  const float* sigma  = (const float*)d_in[1];  // (16,)
  const float* alpha  = (const float*)d_in[2];  // (16,)
  const float* images = (const float*)d_in[3];  // (16,32,32,3)
  float* out = (float*)d_out;                   // (16,3072)

  float* ws     = (float*)d_ws;
  float* fimg   = ws;                    // 16*3072 floats
  float* xnorm  = fimg + NB * IMG_DIM;   // 16
  float* den    = xnorm + NB;            // 16
  float* pdfbuf = den + NB;              // 16*10000 floats

  prep_kernel<<<NB, 256, 0, stream>>>(db, sigma, alpha, images, fimg, xnorm, den);
  dist_pdf_kernel<<<NDB / 16, 256, 0, stream>>>(db, sigma, alpha, fimg, xnorm,
                                                pdfbuf, den);
  wsum_kernel<<<IMG_DIM / 16, 256, 0, stream>>>(db, pdfbuf, den, out);
}